// PointPillarScatter_spa_59115929862334
// MI455X (gfx1250) — compile-verified
//
#include <hip/hip_runtime.h>
#include <math.h>

typedef __attribute__((ext_vector_type(2))) float v2f;
typedef __attribute__((ext_vector_type(8))) float v8f;

#define H_DIM 496
#define W_DIM 432
#define C_DIM 64
#define B_DIM 2
#define N_PTS 8192
#define HWSZ  (H_DIM * W_DIM)
#define EPSLN 1e-5f

// Async global->LDS copy (CDNA5 TDM-adjacent path, tracked by ASYNCcnt).
// lds_off = LDS byte address (low 32 bits of generic shared pointer),
// gaddr   = 64-bit global address (VGPR pair).
__device__ __forceinline__ void async_copy_b128(unsigned lds_off, const float* gaddr) {
  asm volatile("global_load_async_to_lds_b128 %0, %1, off"
               :: "v"(lds_off), "v"(gaddr) : "memory");
}

// ------------------------------------------------------------------
// Kernel 0: zero the (B, C, H, W) output (everything not scattered is 0)
// ------------------------------------------------------------------
__global__ __launch_bounds__(256) void k_zero(float* __restrict__ out, long long n4) {
  long long i = (long long)blockIdx.x * blockDim.x + threadIdx.x;
  if (i < n4) {
    float4 z; z.x = 0.f; z.y = 0.f; z.z = 0.f; z.w = 0.f;
    ((float4*)out)[i] = z;
  }
}

// ------------------------------------------------------------------
// Kernel 1: per-point  g = p + rope(p)  -> LN -> Q,K,V projections.
// One wave per point row; lane handles channels (l, l+32) (RoPE pair).
// ------------------------------------------------------------------
__global__ __launch_bounds__(256) void k_rope_ln_qkv(
    const float* __restrict__ pf, const int* __restrict__ vidx,
    const float* __restrict__ qw, const float* __restrict__ qb,
    const float* __restrict__ kw, const float* __restrict__ kb,
    const float* __restrict__ vw, const float* __restrict__ vb,
    const float* __restrict__ n1g, const float* __restrict__ n1b,
    float* __restrict__ g, float* __restrict__ Q,
    float* __restrict__ K, float* __restrict__ V) {
  __shared__ float sh[8][64];
  const int wid  = threadIdx.x >> 5;
  const int lane = threadIdx.x & 31;
  const long long row = (long long)blockIdx.x * 8 + wid;   // [0, B*N)

  const float* prow = pf + row * C_DIM;
  float p1 = prow[lane];
  float p2 = prow[lane + 32];

  const int pos = vidx[row];
  const int hh = pos / W_DIM;
  const int ww = pos - hh * W_DIM;
  const float hf = -1.f + 2.f * (float)hh / (float)(H_DIM - 1);
  const float wf = -1.f + 2.f * (float)ww / (float)(W_DIM - 1);
  // theta = 10000^(-lane/32) = exp(-lane/32 * ln 1e4)
  const float theta = __expf(-(float)lane * (9.210340371976184f / 32.f));
  const float hs = __sinf(hf * theta), hc = __cosf(hf * theta);
  const float wsn = __sinf(wf * theta), wc = __cosf(wf * theta);

  const float g1 = p1 + p1 * hc * wc - p2 * hs * wsn;
  const float g2 = p2 + p1 * hs * wsn + p2 * hc * wc;
  float* grow = g + row * C_DIM;
  grow[lane] = g1;
  grow[lane + 32] = g2;

  // LayerNorm over 64 channels
  float s = g1 + g2, s2 = g1 * g1 + g2 * g2;
  #pragma unroll
  for (int o = 16; o >= 1; o >>= 1) { s += __shfl_xor(s, o); s2 += __shfl_xor(s2, o); }
  const float m    = s * (1.f / 64.f);
  const float var  = s2 * (1.f / 64.f) - m * m;
  const float rstd = rsqrtf(var + EPSLN);
  const float gn1 = (g1 - m) * rstd * n1g[lane] + n1b[lane];
  const float gn2 = (g2 - m) * rstd * n1g[lane + 32] + n1b[lane + 32];
  sh[wid][lane] = gn1;
  sh[wid][lane + 32] = gn2;
  __syncthreads();

  // lane computes output channels j = lane and j = lane+32 of Q, K, V
  float aq1 = qb[lane], aq2 = qb[lane + 32];
  float ak1 = kb[lane], ak2 = kb[lane + 32];
  float av1 = vb[lane], av2 = vb[lane + 32];
  const float* q1 = qw + lane * 64;       const float* q2 = qw + (lane + 32) * 64;
  const float* k1 = kw + lane * 64;       const float* k2 = kw + (lane + 32) * 64;
  const float* v1 = vw + lane * 64;       const float* v2 = vw + (lane + 32) * 64;
  #pragma unroll 8
  for (int c = 0; c < 64; ++c) {
    const float x = sh[wid][c];
    aq1 = fmaf(x, q1[c], aq1); aq2 = fmaf(x, q2[c], aq2);
    ak1 = fmaf(x, k1[c], ak1); ak2 = fmaf(x, k2[c], ak2);
    av1 = fmaf(x, v1[c], av1); av2 = fmaf(x, v2[c], av2);
  }
  float* qr = Q + row * C_DIM; qr[lane] = aq1; qr[lane + 32] = aq2;
  float* kr = K + row * C_DIM; kr[lane] = ak1; kr[lane + 32] = ak2;
  float* vr = V + row * C_DIM; vr[lane] = av1; vr[lane + 32] = av2;
}

// ------------------------------------------------------------------
// Kernel 2: flash attention, fp32 WMMA (V_WMMA_F32_16X16X4_F32).
// WG = 256 threads = 8 waves; wave owns 16 query rows; streams 512
// 16-wide key tiles, double-buffered in LDS via async global->LDS
// loads (ASYNCcnt). Computes S^T = K*Q^T so softmax (over keys)
// reduces across VGPRs + one xor-16 shuffle.
// Output: att = softmax(QK^T)V + g.
// ------------------------------------------------------------------
__global__ __launch_bounds__(256) void k_flash(
    const float* __restrict__ Q, const float* __restrict__ Kmat,
    const float* __restrict__ Vmat, const float* __restrict__ g,
    float* __restrict__ att) {
  // +4 dword row pad: bank-conflict-free frag loads, rows stay 16B aligned
  __shared__ float lK[2][16 * 68];
  __shared__ float lV[2][16 * 68];

  const int lane  = threadIdx.x & 31;
  const int wid   = threadIdx.x >> 5;
  const int qblk  = blockIdx.x & 63;   // 64 query blocks of 128 rows
  const int b     = blockIdx.x >> 6;
  const int l15   = lane & 15;
  const int khalf = (lane >= 16) ? 2 : 0;   // K-dim sub-pair selector (A/B frag layout)
  const int q0    = qblk * 128 + wid * 16;

  const float* Qb = Q    + (long long)b * N_PTS * C_DIM;
  const float* Kb = Kmat + (long long)b * N_PTS * C_DIM;
  const float* Vb = Vmat + (long long)b * N_PTS * C_DIM;

  // Preload Q^T B-fragments for this wave's 16 queries (reused for all key tiles)
  v2f bq[16];
  {
    const float* qrow = Qb + (long long)(q0 + l15) * C_DIM + khalf;
    #pragma unroll
    for (int c = 0; c < 16; ++c) bq[c] = *(const v2f*)(qrow + 4 * c);
  }

  v8f acc[4]; // O^T accumulators: feature tiles f=0..3 (feat M-dim, query N-dim)
  #pragma unroll
  for (int f = 0; f < 4; ++f) acc[f] = (v8f)0.f;
  float m_run = -3.0e38f;
  float l_run = 0.f;

  // staging: 256 threads x 16B = one 16x64 f32 tile per matrix
  const int srow = threadIdx.x >> 4;
  const int scol = (threadIdx.x & 15) * 4;
  const unsigned lkoff0 = (unsigned)(size_t)&lK[0][srow * 68 + scol];
  const unsigned lkoff1 = (unsigned)(size_t)&lK[1][srow * 68 + scol];
  const unsigned lvoff0 = (unsigned)(size_t)&lV[0][srow * 68 + scol];
  const unsigned lvoff1 = (unsigned)(size_t)&lV[1][srow * 68 + scol];

  // prologue: kick off tile 0 into buffer 0
  async_copy_b128(lkoff0, Kb + (long long)srow * C_DIM + scol);
  async_copy_b128(lvoff0, Vb + (long long)srow * C_DIM + scol);

  for (int j = 0; j < N_PTS / 16; ++j) {
    const int cur = j & 1;
    // all waves are done reading the buffer tile j+1 will overwrite
    __syncthreads();
    if (j + 1 < N_PTS / 16) {
      const long long r = (long long)((j + 1) * 16 + srow) * C_DIM + scol;
      async_copy_b128(cur ? lkoff0 : lkoff1, Kb + r);
      async_copy_b128(cur ? lvoff0 : lvoff1, Vb + r);
      // 4 outstanding; oldest 2 (= tile j's K+V) done when <= 2 remain
      asm volatile("s_wait_asynccnt 0x2" ::: "memory");
    } else {
      asm volatile("s_wait_asynccnt 0x0" ::: "memory");
    }
    __syncthreads();   // tile j visible to all waves

    const float* __restrict__ tk = lK[cur];
    const float* __restrict__ tv = lV[cur];

    // S^T tile (16 keys x 16 queries) = K_tile(16x64) x Q^T(64x16), 16 chained WMMAs
    v8f s = (v8f)0.f;
    #pragma unroll
    for (int c = 0; c < 16; ++c) {
      const v2f ak = *(const v2f*)(&tk[l15 * 68 + 4 * c + khalf]);
      s = __builtin_amdgcn_wmma_f32_16x16x4_f32(false, ak, false, bq[c],
                                                (short)0, s, false, false);
    }

    // online softmax; per-lane stats are for query (lane&15), mirrored in both halves
    float tm = s[0];
    #pragma unroll
    for (int r = 1; r < 8; ++r) tm = fmaxf(tm, s[r]);
    tm = fmaxf(tm, __shfl_xor(tm, 16));
    const float m_new = fmaxf(m_run, tm);
    const float scl   = __expf(m_run - m_new);
    v8f p;
    float ts = 0.f;
    #pragma unroll
    for (int r = 0; r < 8; ++r) { p[r] = __expf(s[r] - m_new); ts += p[r]; }
    ts += __shfl_xor(ts, 16);
    l_run = l_run * scl + ts;
    m_run = m_new;
    #pragma unroll
    for (int f = 0; f < 4; ++f) acc[f] *= scl;

    // Re-lay P^T (key=M in VGPRs, query=N in lanes) into B-fragments (key chunks of 4)
    v2f bp[4];
    #pragma unroll
    for (int c = 0; c < 4; ++c) {
      const int kx0 = 4 * c, kx1 = 4 * c + 2, ky0 = 4 * c + 1, ky1 = 4 * c + 3;
      const float xa = __shfl(p[kx0 & 7], l15 + ((kx0 & 8) ? 16 : 0));
      const float xb = __shfl(p[kx1 & 7], l15 + ((kx1 & 8) ? 16 : 0));
      const float ya = __shfl(p[ky0 & 7], l15 + ((ky0 & 8) ? 16 : 0));
      const float yb = __shfl(p[ky1 & 7], l15 + ((ky1 & 8) ? 16 : 0));
      bp[c].x = (lane < 16) ? xa : xb;
      bp[c].y = (lane < 16) ? ya : yb;
    }

    // O^T += V^T(64x16) x P^T(16x16): 4 feature tiles x 4 key chunks
    #pragma unroll
    for (int f = 0; f < 4; ++f) {
      #pragma unroll
      for (int c = 0; c < 4; ++c) {
        v2f av;
        av.x = tv[(4 * c + khalf)     * 68 + 16 * f + l15];
        av.y = tv[(4 * c + khalf + 1) * 68 + 16 * f + l15];
        acc[f] = __builtin_amdgcn_wmma_f32_16x16x4_f32(false, av, false, bp[c],
                                                       (short)0, acc[f], false, false);
      }
    }
  }

  // epilogue: att[q][feat] = O[q][feat]/l + g[q][feat]
  const float rdiv = 1.f / l_run;
  const float* grow = g   + (long long)b * N_PTS * C_DIM;
  float*       arow = att + (long long)b * N_PTS * C_DIM;
  const long long qrow = (long long)(q0 + l15) * C_DIM;
  const int fo = (lane >= 16) ? 8 : 0;
  #pragma unroll
  for (int r = 0; r < 8; ++r) {
    #pragma unroll
    for (int f = 0; f < 4; ++f) {
      const int feat = 16 * f + r + fo;
      arow[qrow + feat] = acc[f][r] * rdiv + grow[qrow + feat];
    }
  }
}

// ------------------------------------------------------------------
// Kernel 3: LN2 -> FFN (exact GELU) -> residual -> scatter to (C,H,W)
// ------------------------------------------------------------------
__global__ __launch_bounds__(256) void k_ffn_scatter(
    const float* __restrict__ att, const int* __restrict__ vidx,
    const float* __restrict__ w1, const float* __restrict__ b1,
    const float* __restrict__ w2, const float* __restrict__ b2,
    const float* __restrict__ n2g, const float* __restrict__ n2b,
    float* __restrict__ out) {
  __shared__ float sh[8][64];
  const int wid  = threadIdx.x >> 5;
  const int lane = threadIdx.x & 31;
  const long long row = (long long)blockIdx.x * 8 + wid;
  const long long b   = row / N_PTS;

  const float* arow = att + row * C_DIM;
  const float a1 = arow[lane], a2 = arow[lane + 32];
  float s = a1 + a2, s2 = a1 * a1 + a2 * a2;
  #pragma unroll
  for (int o = 16; o >= 1; o >>= 1) { s += __shfl_xor(s, o); s2 += __shfl_xor(s2, o); }
  const float m    = s * (1.f / 64.f);
  const float var  = s2 * (1.f / 64.f) - m * m;
  const float rstd = rsqrtf(var + EPSLN);
  sh[wid][lane]      = (a1 - m) * rstd * n2g[lane]      + n2b[lane];
  sh[wid][lane + 32] = (a2 - m) * rstd * n2g[lane + 32] + n2b[lane + 32];
  __syncthreads();

  float h1 = b1[lane], h2 = b1[lane + 32];
  const float* wa = w1 + lane * 64;       const float* wb = w1 + (lane + 32) * 64;
  #pragma unroll 8
  for (int c = 0; c < 64; ++c) {
    const float x = sh[wid][c];
    h1 = fmaf(x, wa[c], h1); h2 = fmaf(x, wb[c], h2);
  }
  h1 = 0.5f * h1 * (1.f + erff(h1 * 0.7071067811865475f));
  h2 = 0.5f * h2 * (1.f + erff(h2 * 0.7071067811865475f));
  __syncthreads();
  sh[wid][lane] = h1; sh[wid][lane + 32] = h2;
  __syncthreads();

  float u1 = b2[lane] + a1, u2 = b2[lane + 32] + a2;
  const float* ua = w2 + lane * 64;       const float* ub = w2 + (lane + 32) * 64;
  #pragma unroll 8
  for (int c = 0; c < 64; ++c) {
    const float x = sh[wid][c];
    u1 = fmaf(x, ua[c], u1); u2 = fmaf(x, ub[c], u2);
  }

  const int pos = vidx[row];
  float* ob = out + b * (long long)C_DIM * HWSZ;
  ob[(long long)lane        * HWSZ + pos] = u1;
  ob[(long long)(lane + 32) * HWSZ + pos] = u2;
}

// ------------------------------------------------------------------
extern "C" void kernel_launch(void* const* d_in, const int* in_sizes, int n_in,
                              void* d_out, int out_size, void* d_ws, size_t ws_size,
                              hipStream_t stream) {
  (void)in_sizes; (void)n_in; (void)out_size; (void)ws_size;
  const float* pf   = (const float*)d_in[0];
  const int*   vidx = (const int*)  d_in[1];
  const float* qw   = (const float*)d_in[2];
  const float* qb   = (const float*)d_in[3];
  const float* kw   = (const float*)d_in[4];
  const float* kb   = (const float*)d_in[5];
  const float* vw   = (const float*)d_in[6];
  const float* vb   = (const float*)d_in[7];
  const float* w1   = (const float*)d_in[8];
  const float* b1   = (const float*)d_in[9];
  const float* w2   = (const float*)d_in[10];
  const float* b2   = (const float*)d_in[11];
  const float* n1g  = (const float*)d_in[12];
  const float* n1b  = (const float*)d_in[13];
  const float* n2g  = (const float*)d_in[14];
  const float* n2b  = (const float*)d_in[15];
  float* out = (float*)d_out;
  float* ws  = (float*)d_ws;

  const long long rc = (long long)B_DIM * N_PTS * C_DIM;  // 1,048,576 floats
  float* g   = ws;
  float* Q   = ws + 1 * rc;
  float* K   = ws + 2 * rc;
  float* V   = ws + 3 * rc;
  float* att = ws + 4 * rc;

  const long long n4 = (long long)B_DIM * C_DIM * HWSZ / 4;
  k_zero<<<(int)((n4 + 255) / 256), 256, 0, stream>>>(out, n4);
  k_rope_ln_qkv<<<B_DIM * N_PTS / 8, 256, 0, stream>>>(pf, vidx, qw, qb, kw, kb,
                                                       vw, vb, n1g, n1b, g, Q, K, V);
  k_flash<<<B_DIM * 64, 256, 0, stream>>>(Q, K, V, g, att);
  k_ffn_scatter<<<B_DIM * N_PTS / 8, 256, 0, stream>>>(att, vidx, w1, b1, w2, b2,
                                                       n2g, n2b, out);
}